// hierESN_22771916603669
// MI455X (gfx1250) — compile-verified
//
#include <hip/hip_runtime.h>
#include <math.h>

#define T_STEPS 2048
#define N_INPUT 4096
#define N_RES   2000
#define ALPHA_C 0.5f
#define SCAN_BLOCKS 64

typedef float v2f __attribute__((ext_vector_type(2)));
typedef float v8f __attribute__((ext_vector_type(8)));

// ---------------------------------------------------------------------------
// Zero the double-buffered reservoir states and the grid-barrier counters.
// Runs once per kernel_launch so graph replays are deterministic.
// ---------------------------------------------------------------------------
__global__ void esn_init(float* v1buf, float* v2buf, unsigned* counters) {
    int i = threadIdx.x;
    for (int j = i; j < 2 * 2048; j += blockDim.x) {
        v1buf[j] = 0.f;
        v2buf[j] = 0.f;
    }
    if (i < 8) counters[i] = 0u;
}

// ---------------------------------------------------------------------------
// C[M,N] = A[M,K] @ B[K,N], fp32, via V_WMMA_F32_16X16X4_F32.
// Each wave computes a 64x16 strip of C (4 stacked 16x16 m-tiles) so one B
// fragment feeds 4 WMMAs: 4 A b64-loads + 2 B b32-loads per 4 WMMAs.
// Distinct fragment registers per m-tile let all loads issue as one clause
// before the WMMA chain (staggered s_wait_loadcnt instead of serialized).
// Requires M%64==0, N%16==0, K%4==0 (true for all calls here).
// 8 waves (256 threads) per block, one strip per wave.
// ---------------------------------------------------------------------------
__global__ void __launch_bounds__(256)
wmma_gemm_f32(const float* __restrict__ A, int lda,
              const float* __restrict__ B, int ldb,
              float* __restrict__ C, int ldc,
              int M, int N, int K) {
    const int wave = threadIdx.x >> 5;
    const int lane = threadIdx.x & 31;
    const int tiles_n  = N >> 4;
    const int strips_m = M >> 6;
    const int job = blockIdx.x * 8 + wave;
    if (job >= strips_m * tiles_n) return;    // whole wave exits together
    const int m0 = (job / tiles_n) << 6;
    const int n0 = (job % tiles_n) << 4;

    const int lrow  = lane & 15;              // M index (A) / N index (B,C)
    const int khalf = (lane >> 4) << 1;       // lanes 0-15 -> K+0/1, 16-31 -> K+2/3

    const float* __restrict__ a0 = A + (size_t)(m0 + lrow) * lda + khalf;
    const size_t astep = (size_t)16 * lda;    // next 16-row m-tile
    const float* __restrict__ bc = B + (size_t)khalf * ldb + (n0 + lrow);

    v8f c0 = {0.f,0.f,0.f,0.f,0.f,0.f,0.f,0.f};
    v8f c1 = c0, c2 = c0, c3 = c0;

#pragma unroll 2
    for (int k = 0; k < K; k += 4) {
        // issue all fragment loads up front (independent registers)
        v2f b, f0, f1, f2, f3;
        b[0]  = bc[(size_t)k * ldb];          // (k+khalf,   n0+lrow)
        b[1]  = bc[(size_t)(k + 1) * ldb];    // (k+khalf+1, n0+lrow)
        f0[0] = a0[k];
        f0[1] = a0[k + 1];
        f1[0] = a0[astep + k];
        f1[1] = a0[astep + k + 1];
        f2[0] = a0[2 * astep + k];
        f2[1] = a0[2 * astep + k + 1];
        f3[0] = a0[3 * astep + k];
        f3[1] = a0[3 * astep + k + 1];
        c0 = __builtin_amdgcn_wmma_f32_16x16x4_f32(false, f0, false, b, (short)0, c0, false, false);
        c1 = __builtin_amdgcn_wmma_f32_16x16x4_f32(false, f1, false, b, (short)0, c1, false, false);
        c2 = __builtin_amdgcn_wmma_f32_16x16x4_f32(false, f2, false, b, (short)0, c2, false, false);
        c3 = __builtin_amdgcn_wmma_f32_16x16x4_f32(false, f3, false, b, (short)0, c3, false, false);
    }

    // C/D layout per tile: VGPR r -> row base+r (lanes 0-15) or base+8+r
    // (lanes 16-31), col n0+lane%16.
    float* crow = C + (size_t)(m0 + ((lane >> 4) << 3)) * ldc + (n0 + lrow);
#pragma unroll
    for (int r = 0; r < 8; ++r) crow[(size_t)r * ldc] = c0[r];
    crow += (size_t)16 * ldc;
#pragma unroll
    for (int r = 0; r < 8; ++r) crow[(size_t)r * ldc] = c1[r];
    crow += (size_t)16 * ldc;
#pragma unroll
    for (int r = 0; r < 8; ++r) crow[(size_t)r * ldc] = c2[r];
    crow += (size_t)16 * ldc;
#pragma unroll
    for (int r = 0; r < 8; ++r) crow[(size_t)r * ldc] = c3[r];
}

// ---------------------------------------------------------------------------
// Persistent sequential scan:  v <- (1-a)*v + a*tanh(v @ W + U[t]),
// storing each state into Vout (row stride ldv). Double-buffered state in
// global memory, one cumulative-count grid barrier per time step.
// Grid must be exactly SCAN_BLOCKS blocks of 256 threads.
// Each block owns 32 columns; each column's dot is split over 8 k-slices.
// ---------------------------------------------------------------------------
__global__ void __launch_bounds__(256)
esn_scan(const float* __restrict__ W,     // [N,N] row-major
         const float* __restrict__ U,     // [T,N]
         float* __restrict__ Vout, int ldv,
         float* __restrict__ vbuf,        // [2][2048]
         unsigned* __restrict__ counter,
         int N, int T, float alpha) {
    __shared__ float vlds[2048];
    __shared__ float red[256];

    const int tid  = threadIdx.x;
    const int colk = tid & 31;
    const int ksl  = tid >> 5;
    const int col  = blockIdx.x * 32 + colk;
    const int kchunk = (N + 7) >> 3;          // 250 for N=2000
    const int k0 = ksl * kchunk;
    const int k1 = (k0 + kchunk < N) ? (k0 + kchunk) : N;
    const unsigned nb = gridDim.x;

    for (int t = 0; t < T; ++t) {
        const float* vold = vbuf + ((t & 1) << 11);
        float*       vnew = vbuf + (((t + 1) & 1) << 11);

        // stage current state in LDS
        for (int i = tid; i < N; i += 256) vlds[i] = vold[i];
        __syncthreads();

        // partial dot over this thread's k-slice (coalesced W reads)
        float acc = 0.f;
        if (col < N) {
            const float* wc = W + col;
            for (int k = k0; k < k1; ++k)
                acc = fmaf(vlds[k], wc[(size_t)k * N], acc);
        }
        red[tid] = acc;
        __syncthreads();

        // reduce 8 slices, apply leaky tanh update, publish
        if (tid < 32) {
            float s = red[tid];
#pragma unroll
            for (int j = 1; j < 8; ++j) s += red[tid + (j << 5)];
            if (col < N) {
                float pre = s + U[(size_t)t * N + col];
                float vn  = (1.f - alpha) * vlds[col] + alpha * tanhf(pre);
                vnew[col] = vn;
                Vout[(size_t)t * ldv + col] = vn;
            }
        }

        // ---- grid barrier (release -> signal -> acquire) ----
        __syncthreads();
        __threadfence();
        __syncthreads();
        if (tid == 0) {
            atomicAdd(counter, 1u);
            const unsigned target = (unsigned)(t + 1) * nb;
            while (__hip_atomic_load(counter, __ATOMIC_ACQUIRE,
                                     __HIP_MEMORY_SCOPE_AGENT) < target) {
                __builtin_amdgcn_s_sleep(1);
            }
        }
        __syncthreads();
        __threadfence();
    }
}

// ---------------------------------------------------------------------------
// Launch: init, GEMM1, scan1 (writes V1 into out cols [0,2000)),
//         GEMM2 (A = V1 read from out with lda=4000), scan2 (cols [2000,4000)).
// ---------------------------------------------------------------------------
extern "C" void kernel_launch(void* const* d_in, const int* in_sizes, int n_in,
                              void* d_out, int out_size, void* d_ws, size_t ws_size,
                              hipStream_t stream) {
    const float* x    = (const float*)d_in[0];   // [2048, 4096]
    const float* Win1 = (const float*)d_in[1];   // [4096, 2000]
    const float* W1   = (const float*)d_in[2];   // [2000, 2000]
    const float* Win2 = (const float*)d_in[3];   // [2000, 2000]
    const float* W2   = (const float*)d_in[4];   // [2000, 2000]
    float* out = (float*)d_out;                  // [2048, 4000]

    float* U1 = (float*)d_ws;                               // T*N_RES floats
    float* U2 = U1 + (size_t)T_STEPS * N_RES;               // T*N_RES floats
    float* v1buf = U2 + (size_t)T_STEPS * N_RES;            // 2*2048
    float* v2buf = v1buf + 2 * 2048;                        // 2*2048
    unsigned* counters = (unsigned*)(v2buf + 2 * 2048);     // 8 u32

    esn_init<<<1, 256, 0, stream>>>(v1buf, v2buf, counters);

    // U1 = X @ Win1 : M=2048, N=2000, K=4096 -> (2048/64)*(2000/16)=4000 wave
    // jobs -> 500 blocks of 8 waves
    wmma_gemm_f32<<<500, 256, 0, stream>>>(x, N_INPUT, Win1, N_RES,
                                           U1, N_RES,
                                           T_STEPS, N_RES, N_INPUT);

    // scan reservoir 1, states go to out[:, 0:2000]
    esn_scan<<<SCAN_BLOCKS, 256, 0, stream>>>(W1, U1, out, 2 * N_RES,
                                              v1buf, counters + 0,
                                              N_RES, T_STEPS, ALPHA_C);

    // U2 = V1 @ Win2 : A is out with row stride 4000, K=2000
    wmma_gemm_f32<<<500, 256, 0, stream>>>(out, 2 * N_RES, Win2, N_RES,
                                           U2, N_RES,
                                           T_STEPS, N_RES, N_RES);

    // scan reservoir 2, states go to out[:, 2000:4000]
    esn_scan<<<SCAN_BLOCKS, 256, 0, stream>>>(W2, U2, out + N_RES, 2 * N_RES,
                                              v2buf, counters + 1,
                                              N_RES, T_STEPS, ALPHA_C);
}